// NLPFeaturesRNN_66614942761206
// MI455X (gfx1250) — compile-verified
//
#include <hip/hip_runtime.h>
#include <math.h>

// dims
#define B_   512
#define T_   70
#define E_   300
#define H_   64
#define NF_  16
#define NH_  64
#define FCIN 9984

typedef __attribute__((ext_vector_type(2))) float v2f;
typedef __attribute__((ext_vector_type(8))) float v8f;

__device__ __forceinline__ v8f wmma4(v2f a, v2f b, v8f c) {
  // D = A(16x4) * B(4x16) + C(16x16), fp32 end to end
  return __builtin_amdgcn_wmma_f32_16x16x4_f32(false, a, false, b, (short)0, c,
                                               false, false);
}

__device__ __forceinline__ float sigf(float x) { return 1.f / (1.f + expf(-x)); }

// ---------------------------------------------------------------------------
// Tensor Data Mover: dense 2D tile (rows x cols fp32, row stride in elems)
// from global memory into LDS (row-major packed). Tracked by TENSORcnt.
// ---------------------------------------------------------------------------
#if __has_builtin(__builtin_amdgcn_tensor_load_to_lds)
#define HAVE_TDM 1
typedef __attribute__((ext_vector_type(4))) unsigned int v4u;
typedef __attribute__((ext_vector_type(4))) int v4i;
typedef __attribute__((ext_vector_type(8))) int v8i;

__device__ __forceinline__ void tdm_load_2d(const float* gsrc, unsigned ldsByte,
                                            int rows, int cols, int strideElems) {
  unsigned long long ga = (unsigned long long)gsrc;
  unsigned galo = (unsigned)__builtin_amdgcn_readfirstlane((int)(ga & 0xffffffffull));
  unsigned gahi = (unsigned)__builtin_amdgcn_readfirstlane((int)(ga >> 32));
  unsigned lb   = (unsigned)__builtin_amdgcn_readfirstlane((int)ldsByte);
  // D# group 0: count=1 | lds_addr | global_addr[56:0] | type=2
  v4u g0;
  g0.x = 1u;
  g0.y = lb;
  g0.z = galo;
  g0.w = (gahi & 0x01ffffffu) | (2u << 30);
  // D# group 1: data_size=4B, tensor=cols x rows, tile=cols x rows, dim0 stride
  unsigned td0 = (unsigned)cols;
  unsigned td1 = (unsigned)rows;
  unsigned long long st0 = (unsigned long long)(unsigned)strideElems;
  v8i g1;
  g1[0] = (int)0x00020000u;                             // data_size = 2 (4 bytes)
  g1[1] = (int)((td0 & 0xffffu) << 16);                 // tensor_dim0[15:0]
  g1[2] = (int)(((td1 & 0xffffu) << 16) | (td0 >> 16)); // tensor_dim1lo | dim0hi
  g1[3] = (int)(((unsigned)cols << 16) | (td1 >> 16));  // tile_dim0 | dim1hi
  g1[4] = (int)(rows & 0xffff);                         // tile_dim1 (tile_dim2=0)
  g1[5] = (int)(st0 & 0xffffffffull);                   // tensor_dim0_stride lo
  g1[6] = (int)((st0 >> 32) & 0xffffull);               // dim0_stride hi (dim1_stride=0)
  g1[7] = 0;
  v4i z4 = {0, 0, 0, 0};
#if __has_include(<hip/amd_detail/amd_gfx1250_TDM.h>)
  v8i z8 = {0, 0, 0, 0, 0, 0, 0, 0};
  __builtin_amdgcn_tensor_load_to_lds(g0, g1, z4, z4, z8, 0);
#else
  __builtin_amdgcn_tensor_load_to_lds(g0, g1, z4, z4, 0);
#endif
}
#else
#define HAVE_TDM 0
#endif

// ---------------------------------------------------------------------------
// W [N,64] -> WT [64,N]   (so recurrent B-fragments are lane-coalesced)
// ---------------------------------------------------------------------------
__global__ void __launch_bounds__(256)
transpose64_kernel(const float* __restrict__ W, float* __restrict__ WT, int N) {
  int idx = blockIdx.x * 256 + threadIdx.x;
  if (idx < N * 64) {
    int n = idx / 64, k = idx % 64;
    WT[k * N + n] = W[idx];
  }
}

// ---------------------------------------------------------------------------
// Generic (gather +) GEMM for input projections.
//   out_dir[m][n] = sum_k A[m][k] * W_dir[n][k] + bias1[n] (+ bias2[n])
// One block = one 16-row M tile staged in LDS (via TDM when dense),
// 8 waves sweep 2*Nd columns.
// ---------------------------------------------------------------------------
__global__ void __launch_bounds__(256)
proj_gemm_kernel(const float* __restrict__ A_base, const int* __restrict__ gather,
                 int K,
                 const float* __restrict__ Wf, const float* __restrict__ bf1,
                 const float* __restrict__ bf2,
                 const float* __restrict__ Wb, const float* __restrict__ bb1,
                 const float* __restrict__ bb2,
                 int Nd, float* __restrict__ out_f, float* __restrict__ out_b) {
  extern __shared__ float ldsA[];  // 16*K floats
  const int m0 = blockIdx.x * 16;
#if HAVE_TDM
  if (gather == nullptr) {
    if (threadIdx.x < 32)
      tdm_load_2d(A_base + (size_t)m0 * K,
                  (unsigned)(unsigned long long)&ldsA[0], 16, K, K);
    __builtin_amdgcn_s_wait_tensorcnt(0);
  } else
#endif
  {
    for (int e = threadIdx.x; e < 16 * K; e += 256) {
      int r = e / K, c = e - r * K;
      int row = m0 + r;
      long src = gather ? (long)gather[row] : (long)row;
      ldsA[e] = A_base[src * K + c];
    }
  }
  __syncthreads();

  const int wave = threadIdx.x >> 5;
  const int lane = threadIdx.x & 31;
  const int half = lane >> 4;
  const int l16  = lane & 15;
  const int nTiles = (2 * Nd) >> 4;

  for (int tile = wave; tile < nTiles; tile += 8) {
    int cg   = tile << 4;
    int bwd  = (cg >= Nd);
    int col0 = bwd ? (cg - Nd) : cg;
    const float* W  = bwd ? Wb : Wf;
    const float* b1 = bwd ? bb1 : bf1;
    const float* b2 = bwd ? bb2 : bf2;
    float* out      = bwd ? out_b : out_f;

    v8f acc = {};
    const float* arow = ldsA + l16 * K;
    const float* wrow = W + (long)(col0 + l16) * K;
    for (int k = 0; k < K; k += 4) {
      v2f a, b;
      a.x = arow[k + 2 * half];
      a.y = arow[k + 2 * half + 1];
      b.x = wrow[k + 2 * half];
      b.y = wrow[k + 2 * half + 1];
      acc = wmma4(a, b, acc);
    }
    int n = col0 + l16;
    float bv = b1[n] + (b2 ? b2[n] : 0.f);
    for (int r = 0; r < 8; ++r) {
      int row = m0 + r + 8 * half;
      out[(long)row * Nd + n] = acc[r] + bv;
    }
  }
}

// ---------------------------------------------------------------------------
// BiLSTM scan. grid = (B/16, 2 dirs), block = 256 (8 waves).
// h-tile (16x64) in LDS is the WMMA A operand; 16 N-tiles over 8 waves.
// bih+bhh already folded into xg. TDM path double-buffers the per-step
// 16x256 gate tile (row stride T*256): DMA for step t+1 overlaps the
// h@Whh^T WMMAs and gate math of step t.
// ---------------------------------------------------------------------------
__global__ void __launch_bounds__(256)
lstm_scan_kernel(const float* __restrict__ xg_f, const float* __restrict__ xg_b,
                 const float* __restrict__ WT_f, const float* __restrict__ WT_b,
                 float* __restrict__ out /* [B*T,128] */) {
  __shared__ float ldsH[16 * 64];
  __shared__ float ldsC[16 * 64];
  __shared__ float ldsG[16 * 256];
#if HAVE_TDM
  __shared__ float ldsX[2][16 * 256];
#endif
  const int dir = blockIdx.y;
  const int b0  = blockIdx.x * 16;
  const float* xg = dir ? xg_b : xg_f;
  const float* WT = dir ? WT_b : WT_f;  // [64][256]
  for (int e = threadIdx.x; e < 16 * 64; e += 256) { ldsH[e] = 0.f; ldsC[e] = 0.f; }
  __syncthreads();

  const int wave = threadIdx.x >> 5, lane = threadIdx.x & 31;
  const int half = lane >> 4, l16 = lane & 15;
  const int hh = threadIdx.x & 63, brow = threadIdx.x >> 6;

#if HAVE_TDM
  {  // tile for t=0: rows bb at xg[((b0+bb)*T + te0)*256], stride T*256
    const int te0 = dir ? (T_ - 1) : 0;
    if (wave == 0)
      tdm_load_2d(xg + ((long)b0 * T_ + te0) * 256,
                  (unsigned)(unsigned long long)&ldsX[0][0], 16, 256, T_ * 256);
  }
#endif
  for (int t = 0; t < T_; ++t) {
    const int te = dir ? (T_ - 1 - t) : t;
#if HAVE_TDM
    if (t + 1 < T_ && wave == 0) {  // buffer (t+1)&1 free since end of step t-1
      const int tn = dir ? (te - 1) : (te + 1);
      tdm_load_2d(xg + ((long)b0 * T_ + tn) * 256,
                  (unsigned)(unsigned long long)&ldsX[(t + 1) & 1][0],
                  16, 256, T_ * 256);
    }
#endif
    // hg = h @ Whh^T : 16 x 256 (overlaps with in-flight gate-tile DMA)
    for (int tt = 0; tt < 2; ++tt) {
      int col0 = (wave * 2 + tt) << 4;
      v8f acc = {};
      const float* arow = ldsH + l16 * 64;
      for (int k = 0; k < 64; k += 4) {
        v2f a, b;
        a.x = arow[k + 2 * half];
        a.y = arow[k + 2 * half + 1];
        b.x = WT[(k + 2 * half) * 256 + col0 + l16];
        b.y = WT[(k + 2 * half + 1) * 256 + col0 + l16];
        acc = wmma4(a, b, acc);
      }
      for (int r = 0; r < 8; ++r)
        ldsG[(r + 8 * half) * 256 + col0 + l16] = acc[r];
    }
#if HAVE_TDM
    if (t + 1 < T_) __builtin_amdgcn_s_wait_tensorcnt(1);  // tile t done
    else            __builtin_amdgcn_s_wait_tensorcnt(0);
#endif
    __syncthreads();
    // gates
    for (int bb = brow; bb < 16; bb += 4) {
      long mrow = (long)(b0 + bb) * T_ + te;
#if HAVE_TDM
      const float* xr = &ldsX[t & 1][bb * 256];
#else
      const float* xr = xg + mrow * 256;
#endif
      float gi = ldsG[bb * 256 + hh]        + xr[hh];
      float gf = ldsG[bb * 256 + 64 + hh]   + xr[64 + hh];
      float gg = ldsG[bb * 256 + 128 + hh]  + xr[128 + hh];
      float go = ldsG[bb * 256 + 192 + hh]  + xr[192 + hh];
      float c  = sigf(gf) * ldsC[bb * 64 + hh] + sigf(gi) * tanhf(gg);
      float h  = sigf(go) * tanhf(c);
      ldsC[bb * 64 + hh] = c;
      ldsH[bb * 64 + hh] = h;
      out[mrow * 128 + dir * 64 + hh] = h;
#if !HAVE_TDM
      if (t + 1 < T_) {
        long nrow = (long)(b0 + bb) * T_ + (dir ? te - 1 : te + 1);
        __builtin_prefetch(xg + nrow * 256, 0, 1);
      }
#endif
    }
    __syncthreads();
  }
}

// ---------------------------------------------------------------------------
// BiGRU scan. grid = (B/16, 2), block = 256. bhh applied here (needed for r*hn).
// Same TDM double-buffering of the 16x192 gate tile.
// ---------------------------------------------------------------------------
__global__ void __launch_bounds__(256)
gru_scan_kernel(const float* __restrict__ gg_f, const float* __restrict__ gg_b,
                const float* __restrict__ WT_f, const float* __restrict__ WT_b,
                const float* __restrict__ bhh_f, const float* __restrict__ bhh_b,
                float* __restrict__ out /* [B*T,128] */) {
  __shared__ float ldsH[16 * 64];
  __shared__ float ldsG[16 * 192];
#if HAVE_TDM
  __shared__ float ldsX[2][16 * 192];
#endif
  const int dir = blockIdx.y;
  const int b0  = blockIdx.x * 16;
  const float* gg  = dir ? gg_b : gg_f;
  const float* WT  = dir ? WT_b : WT_f;  // [64][192]
  const float* bhh = dir ? bhh_b : bhh_f;
  for (int e = threadIdx.x; e < 16 * 64; e += 256) ldsH[e] = 0.f;
  __syncthreads();

  const int wave = threadIdx.x >> 5, lane = threadIdx.x & 31;
  const int half = lane >> 4, l16 = lane & 15;
  const int hh = threadIdx.x & 63, brow = threadIdx.x >> 6;

#if HAVE_TDM
  {
    const int te0 = dir ? (T_ - 1) : 0;
    if (wave == 0)
      tdm_load_2d(gg + ((long)b0 * T_ + te0) * 192,
                  (unsigned)(unsigned long long)&ldsX[0][0], 16, 192, T_ * 192);
  }
#endif
  for (int t = 0; t < T_; ++t) {
    const int te = dir ? (T_ - 1 - t) : t;
#if HAVE_TDM
    if (t + 1 < T_ && wave == 0) {
      const int tn = dir ? (te - 1) : (te + 1);
      tdm_load_2d(gg + ((long)b0 * T_ + tn) * 192,
                  (unsigned)(unsigned long long)&ldsX[(t + 1) & 1][0],
                  16, 192, T_ * 192);
    }
#endif
    for (int tile = wave; tile < 12; tile += 8) {
      int col0 = tile << 4;
      v8f acc = {};
      const float* arow = ldsH + l16 * 64;
      for (int k = 0; k < 64; k += 4) {
        v2f a, b;
        a.x = arow[k + 2 * half];
        a.y = arow[k + 2 * half + 1];
        b.x = WT[(k + 2 * half) * 192 + col0 + l16];
        b.y = WT[(k + 2 * half + 1) * 192 + col0 + l16];
        acc = wmma4(a, b, acc);
      }
      for (int r = 0; r < 8; ++r)
        ldsG[(r + 8 * half) * 192 + col0 + l16] = acc[r];
    }
#if HAVE_TDM
    if (t + 1 < T_) __builtin_amdgcn_s_wait_tensorcnt(1);
    else            __builtin_amdgcn_s_wait_tensorcnt(0);
#endif
    __syncthreads();
    for (int bb = brow; bb < 16; bb += 4) {
      long mrow = (long)(b0 + bb) * T_ + te;
#if HAVE_TDM
      const float* xr = &ldsX[t & 1][bb * 192];
#else
      const float* xr = gg + mrow * 192;
#endif
      float hr = ldsG[bb * 192 + hh]       + bhh[hh];
      float hz = ldsG[bb * 192 + 64 + hh]  + bhh[64 + hh];
      float hn = ldsG[bb * 192 + 128 + hh] + bhh[128 + hh];
      float r  = sigf(xr[hh] + hr);
      float z  = sigf(xr[64 + hh] + hz);
      float n  = tanhf(xr[128 + hh] + r * hn);
      float h  = (1.f - z) * n + z * ldsH[bb * 64 + hh];
      ldsH[bb * 64 + hh] = h;
      out[mrow * 128 + dir * 64 + hh] = h;
#if !HAVE_TDM
      if (t + 1 < T_) {
        long nrow = (long)(b0 + bb) * T_ + (dir ? te - 1 : te + 1);
        __builtin_prefetch(gg + nrow * 192, 0, 1);
      }
#endif
    }
    __syncthreads();
  }
}

// ---------------------------------------------------------------------------
// Attention (x2) + mean + max pools; writes feats[:,0:1280]. grid=B, block=128.
// ---------------------------------------------------------------------------
__global__ void __launch_bounds__(128)
attn_pool_kernel(const float* __restrict__ lstm_out, const float* __restrict__ gru_out,
                 const float* __restrict__ attL_w, const float* __restrict__ attL_b,
                 const float* __restrict__ attG_w, const float* __restrict__ attG_b,
                 float* __restrict__ feats) {
  __shared__ float sL[T_], sG[T_];
  __shared__ float sumL, sumG;
  const int b = blockIdx.x;
  const int tid = threadIdx.x, wave = tid >> 5, lane = tid & 31;
  const float* xl = lstm_out + (long)b * T_ * 128;
  const float* xgp = gru_out + (long)b * T_ * 128;

  for (int t = wave; t < T_; t += 4) {
    float dl = 0.f, dg = 0.f;
    for (int f = lane; f < 128; f += 32) {
      dl += xl[t * 128 + f] * attL_w[f];
      dg += xgp[t * 128 + f] * attG_w[f];
    }
    for (int off = 16; off; off >>= 1) {
      dl += __shfl_down(dl, off, 32);
      dg += __shfl_down(dg, off, 32);
    }
    if (lane == 0) {
      sL[t] = tanhf(dl + attL_b[t]);
      sG[t] = tanhf(dg + attG_b[t]);
    }
  }
  __syncthreads();
  if (tid == 0) {
    float a = 0.f, c = 0.f;
    for (int t = 0; t < T_; ++t) {
      sL[t] = expf(sL[t]); a += sL[t];
      sG[t] = expf(sG[t]); c += sG[t];
    }
    sumL = a; sumG = c;
  }
  __syncthreads();

  const int f = tid;  // 0..127
  float h1 = 0.f, h2 = 0.f, h3 = 0.f, h4 = -INFINITY;
  const float invL = 1.f / sumL, invG = 1.f / sumG;
  for (int t = 0; t < T_; ++t) {
    float aL = sL[t] * invL + 1e-10f;
    float aG = sG[t] * invG + 1e-10f;
    float vl = xl[t * 128 + f], vg = xgp[t * 128 + f];
    h1 += vl * aL; h2 += vg * aG; h3 += vg; h4 = fmaxf(h4, vg);
  }
  h3 *= (1.f / T_);
  float* fb = feats + (long)b * FCIN;
  fb[f]          = h1;      fb[128 + f]  = h2;
  fb[256 + f]    = h3;      fb[384 + f]  = h4;
  fb[512 + f]    = h1 * h2; fb[640 + f]  = h1 * h3;
  fb[768 + f]    = h1 * h4; fb[896 + f]  = h2 * h3;
  fb[1024 + f]   = h2 * h4; fb[1152 + f] = h3 * h4;
}

// ---------------------------------------------------------------------------
// Per-feature tiny MLPs via WMMA. grid = (B/16, NF), block = 64 (2 waves).
// ---------------------------------------------------------------------------
__global__ void __launch_bounds__(64)
nlp_mlp_kernel(const float* __restrict__ nlp, const float* __restrict__ W0,
               const float* __restrict__ b0, const float* __restrict__ W1,
               const float* __restrict__ b1, float* __restrict__ x1ws,
               float* __restrict__ feats) {
  __shared__ float ldsX0[16 * 64];
  const int bbase = blockIdx.x * 16;
  const int f = blockIdx.y;
  const int tid = threadIdx.x;
  for (int e = tid; e < 16 * 64; e += 64) {
    int bb = e >> 6, o = e & 63;
    float v = nlp[(bbase + bb) * NF_ + f] * W0[f * 64 + o] + b0[f * 64 + o];
    ldsX0[e] = fmaxf(v, 0.f);
  }
  __syncthreads();

  const int wave = tid >> 5, lane = tid & 31, half = lane >> 4, l16 = lane & 15;
  const float* W1f = W1 + (long)f * 64 * 64;  // [o][h]
  for (int tt = 0; tt < 2; ++tt) {
    int col0 = (wave * 2 + tt) << 4;
    v8f acc = {};
    const float* arow = ldsX0 + l16 * 64;
    const float* wrow = W1f + (long)(col0 + l16) * 64;
    for (int k = 0; k < 64; k += 4) {
      v2f a, b;
      a.x = arow[k + 2 * half];
      a.y = arow[k + 2 * half + 1];
      b.x = wrow[k + 2 * half];
      b.y = wrow[k + 2 * half + 1];
      acc = wmma4(a, b, acc);
    }
    for (int r = 0; r < 8; ++r) {
      int bb = r + 8 * half, o = col0 + l16;
      float v = fmaxf(acc[r] + b1[f * 64 + o], 0.f);
      x1ws[((long)(bbase + bb) * NF_ + f) * 64 + o] = v;
      feats[(long)(bbase + bb) * FCIN + 1280 + f * 64 + o] = v;
    }
  }
}

// ---------------------------------------------------------------------------
// Pairwise products: feats[:, 2304:9984]. grid=B, block=256.
// ---------------------------------------------------------------------------
__global__ void __launch_bounds__(256)
pairwise_kernel(const float* __restrict__ x1ws, float* __restrict__ feats) {
  const int b = blockIdx.x;
  const float* x1 = x1ws + (long)b * NF_ * NH_;
  float* fb = feats + (long)b * FCIN + 2304;
  for (int idx = threadIdx.x; idx < 120 * 64; idx += 256) {
    int p = idx >> 6, o = idx & 63;
    int i = 0, rem = p;
    while (rem >= NF_ - 1 - i) { rem -= NF_ - 1 - i; ++i; }
    int j = i + 1 + rem;
    fb[idx] = x1[i * 64 + o] * x1[j * 64 + o];
  }
}

// ---------------------------------------------------------------------------
// Head: relu(feats @ fcW^T + fcb) @ outW^T + outb. grid=B/16, block=128.
// TDM path: double-buffered 16x256 A-tiles, DMA for chunk c+1 overlapped
// with WMMA on chunk c (wait tensorcnt<=1 keeps one DMA in flight).
// ---------------------------------------------------------------------------
__global__ void __launch_bounds__(128)
head_kernel(const float* __restrict__ feats, const float* __restrict__ fcW,
            const float* __restrict__ fcb, const float* __restrict__ outW,
            const float* __restrict__ outb, float* __restrict__ out) {
  __shared__ float ldsA[2][16 * 256];
  __shared__ float ldsO[16 * 64];
  const int m0 = blockIdx.x * 16;
  const int tid = threadIdx.x, wave = tid >> 5, lane = tid & 31;
  const int half = lane >> 4, l16 = lane & 15;
  const int col0 = wave << 4;
  const int NCH = FCIN / 256;  // 39

  v8f acc = {};
#if HAVE_TDM
  if (wave == 0)
    tdm_load_2d(feats + (size_t)m0 * FCIN,
                (unsigned)(unsigned long long)&ldsA[0][0], 16, 256, FCIN);
  for (int c = 0; c < NCH; ++c) {
    if (c + 1 < NCH) {
      if (wave == 0)
        tdm_load_2d(feats + (size_t)m0 * FCIN + (size_t)(c + 1) * 256,
                    (unsigned)(unsigned long long)&ldsA[(c + 1) & 1][0],
                    16, 256, FCIN);
      __builtin_amdgcn_s_wait_tensorcnt(1);  // chunk c complete, c+1 in flight
    } else {
      __builtin_amdgcn_s_wait_tensorcnt(0);
    }
    __syncthreads();
    const float* arow = &ldsA[c & 1][0] + l16 * 256;
    const float* wrow = fcW + (long)(col0 + l16) * FCIN + c * 256;
    for (int k = 0; k < 256; k += 4) {
      v2f a, b;
      a.x = arow[k + 2 * half];
      a.y = arow[k + 2 * half + 1];
      b.x = wrow[k + 2 * half];
      b.y = wrow[k + 2 * half + 1];
      acc = wmma4(a, b, acc);
    }
    __syncthreads();  // reads done before buffer is overwritten next round
  }
#else
  for (int c = 0; c < NCH; ++c) {
    int kc = c * 256;
    __syncthreads();
    for (int e = tid; e < 16 * 256; e += 128)
      ldsA[0][e] = feats[(long)(m0 + (e >> 8)) * FCIN + kc + (e & 255)];
    __syncthreads();
    const float* arow = &ldsA[0][0] + l16 * 256;
    const float* wrow = fcW + (long)(col0 + l16) * FCIN + kc;
    for (int k = 0; k < 256; k += 4) {
      v2f a, b;
      a.x = arow[k + 2 * half];
      a.y = arow[k + 2 * half + 1];
      b.x = wrow[k + 2 * half];
      b.y = wrow[k + 2 * half + 1];
      acc = wmma4(a, b, acc);
    }
  }
#endif
  for (int r = 0; r < 8; ++r)
    ldsO[(r + 8 * half) * 64 + col0 + l16] = fmaxf(acc[r] + fcb[col0 + l16], 0.f);
  __syncthreads();
  if (tid < 16) {
    float s = outb[0];
    for (int o = 0; o < 64; ++o) s += ldsO[tid * 64 + o] * outW[o];
    out[m0 + tid] = s;
  }
}

// ---------------------------------------------------------------------------
extern "C" void kernel_launch(void* const* d_in, const int* in_sizes, int n_in,
                              void* d_out, int out_size, void* d_ws, size_t ws_size,
                              hipStream_t stream) {
  (void)in_sizes; (void)n_in; (void)out_size; (void)ws_size;
  const int*   text  = (const int*)  d_in[0];
  const float* nlp   = (const float*)d_in[1];
  const float* emb   = (const float*)d_in[2];
  const float* lWih_f = (const float*)d_in[3];
  const float* lWhh_f = (const float*)d_in[4];
  const float* lbih_f = (const float*)d_in[5];
  const float* lbhh_f = (const float*)d_in[6];
  const float* lWih_b = (const float*)d_in[7];
  const float* lWhh_b = (const float*)d_in[8];
  const float* lbih_b = (const float*)d_in[9];
  const float* lbhh_b = (const float*)d_in[10];
  const float* gWih_f = (const float*)d_in[11];
  const float* gWhh_f = (const float*)d_in[12];
  const float* gbih_f = (const float*)d_in[13];
  const float* gbhh_f = (const float*)d_in[14];
  const float* gWih_b = (const float*)d_in[15];
  const float* gWhh_b = (const float*)d_in[16];
  const float* gbih_b = (const float*)d_in[17];
  const float* gbhh_b = (const float*)d_in[18];
  const float* attL_w = (const float*)d_in[19];
  const float* attL_b = (const float*)d_in[20];
  const float* attG_w = (const float*)d_in[21];
  const float* attG_b = (const float*)d_in[22];
  const float* W0 = (const float*)d_in[23];
  const float* b0 = (const float*)d_in[24];
  const float* W1 = (const float*)d_in[25];
  const float* b1 = (const float*)d_in[26];
  const float* fcW = (const float*)d_in[27];
  const float* fcb = (const float*)d_in[28];
  const float* outW = (const float*)d_in[29];
  const float* outb = (const float*)d_in[30];
  float* out = (float*)d_out;
  float* ws = (float*)d_ws;

  const int MT = B_ * T_;  // 35840
  // workspace layout (floats), regions aliased by lifetime:
  const size_t XGF   = 0;                 // xg_f [MT,256] -> later gg_f [MT,192] -> later feats [B,9984]
  const size_t XGB   = (size_t)MT * 256;  // xg_b [MT,256] -> later gg_b [MT,192]
  const size_t LSTM  = XGB + (size_t)MT * 256;   // lstm_out [MT,128]
  const size_t GRU   = LSTM + (size_t)MT * 128;  // gru_out  [MT,128]
  const size_t X1    = GRU + (size_t)MT * 128;   // x1 [B,16,64]
  const size_t WT_LF = X1 + (size_t)B_ * NF_ * NH_;
  const size_t WT_LB = WT_LF + 64 * 256;
  const size_t WT_GF = WT_LB + 64 * 256;
  const size_t WT_GB = WT_GF + 64 * 192;

  // 1. weight transposes for recurrent B-operands
  transpose64_kernel<<<(256 * 64 + 255) / 256, 256, 0, stream>>>(lWhh_f, ws + WT_LF, 256);
  transpose64_kernel<<<(256 * 64 + 255) / 256, 256, 0, stream>>>(lWhh_b, ws + WT_LB, 256);
  transpose64_kernel<<<(192 * 64 + 255) / 256, 256, 0, stream>>>(gWhh_f, ws + WT_GF, 192);
  transpose64_kernel<<<(192 * 64 + 255) / 256, 256, 0, stream>>>(gWhh_b, ws + WT_GB, 192);

  // 2. embedding gather + LSTM input projections (bih+bhh folded)
  proj_gemm_kernel<<<MT / 16, 256, 16 * E_ * sizeof(float), stream>>>(
      emb, text, E_, lWih_f, lbih_f, lbhh_f, lWih_b, lbih_b, lbhh_b,
      256, ws + XGF, ws + XGB);

  // 3. BiLSTM scan (TDM double-buffered gate tiles)
  lstm_scan_kernel<<<dim3(B_ / 16, 2), 256, 0, stream>>>(
      ws + XGF, ws + XGB, ws + WT_LF, ws + WT_LB, ws + LSTM);

  // 4. GRU input projections (bih only; bhh applied inside the scan; TDM staging)
  proj_gemm_kernel<<<MT / 16, 256, 16 * 128 * sizeof(float), stream>>>(
      ws + LSTM, nullptr, 128, gWih_f, gbih_f, nullptr, gWih_b, gbih_b, nullptr,
      192, ws + XGF, ws + XGB);

  // 5. BiGRU scan (TDM double-buffered gate tiles)
  gru_scan_kernel<<<dim3(B_ / 16, 2), 256, 0, stream>>>(
      ws + XGF, ws + XGB, ws + WT_GF, ws + WT_GB, gbhh_f, gbhh_b, ws + GRU);

  // 6. attention + pooling -> feats[:,0:1280]   (feats aliases dead xg_f region)
  float* feats = ws + XGF;
  attn_pool_kernel<<<B_, 128, 0, stream>>>(
      ws + LSTM, ws + GRU, attL_w, attL_b, attG_w, attG_b, feats);

  // 7. per-feature MLPs -> x1 + feats[:,1280:2304]
  nlp_mlp_kernel<<<dim3(B_ / 16, NF_), 64, 0, stream>>>(
      nlp, W0, b0, W1, b1, ws + X1, feats);

  // 8. pairwise products -> feats[:,2304:9984]
  pairwise_kernel<<<B_, 256, 0, stream>>>(ws + X1, feats);

  // 9. FC head -> out [B,1]
  head_kernel<<<B_ / 16, 128, 0, stream>>>(feats, fcW, fcb, outW, outb, out);
}